// GATLayer_12807592476895
// MI455X (gfx1250) — compile-verified
//
#include <hip/hip_runtime.h>
#include <math.h>

// CDNA5 / gfx1250: wave32, fp32 WMMA 16x16x4.
typedef __attribute__((ext_vector_type(2))) float v2f;
typedef __attribute__((ext_vector_type(8))) float v8f;

#define WMMA_F32(a, b, c) \
    __builtin_amdgcn_wmma_f32_16x16x4_f32(false, (a), false, (b), (short)0, (c), false, false)

__device__ __forceinline__ float lrelu(float v) { return v > 0.f ? v : 0.2f * v; }
__device__ __forceinline__ v8f splat8(float v) { v8f r = {v, v, v, v, v, v, v, v}; return r; }

// -----------------------------------------------------------------------------
// Kernel 1: fused 3-layer message MLP. One wave32 per 16-row tile of [B*N, 7].
//   x[16x7] -> h1[16x64] -> h2[16x64] -> fx[16x32], all via v_wmma_f32_16x16x4_f32.
// Layouts (per ISA 7.12.2, fp32):
//   A 16x4:  lane(0..15)=M, VGPR0={K0|K2 by half}, VGPR1={K1|K3 by half}
//   B 4x16:  VGPR0 holds rows {K0|K2 by half} striped over N=lane%16, VGPR1={K1|K3}
//   C 16x16: VGPR r -> M = r + 8*half, N = lane%16  (N lane-constant across r,
//            so biases fold into the accumulator splat)
// W1 and the x tile are zero-padded K=7->8 in LDS so every fragment build is
// an unguarded, uniform load (no exec-mask divergence next to WMMA).
// -----------------------------------------------------------------------------
__global__ __launch_bounds__(32)
void gat_mlp_kernel(const float* __restrict__ x,
                    const float* __restrict__ W1, const float* __restrict__ b1,
                    const float* __restrict__ W2, const float* __restrict__ b2,
                    const float* __restrict__ W3, const float* __restrict__ b3,
                    float* __restrict__ fx)
{
    __shared__ float xa[16 * 8];    // x tile, K padded 7 -> 8 with zeros
    __shared__ float w1[8 * 64];    // W1 padded 7 -> 8 rows with zeros
    __shared__ float h[16 * 64];    // hidden activation tile (reused between layers)

    const int lane = threadIdx.x;   // 0..31, exactly one wave32
    const int half = lane >> 4;     // 0 | 1
    const int l16  = lane & 15;
    const int row0 = blockIdx.x * 16;   // over B*N = 8192 rows

    // stage x tile into LDS, zero-padding column 7 (uniform control flow)
    #pragma unroll
    for (int i = 0; i < 4; ++i) {
        int idx = lane * 4 + i;
        int r = idx >> 3, c = idx & 7;
        xa[idx] = (c < 7) ? x[(row0 + r) * 7 + c] : 0.f;
    }
    // stage W1 into LDS, zero-padding row 7
    #pragma unroll
    for (int i = 0; i < 16; ++i) {
        int idx = lane * 16 + i;            // idx = k*64 + n
        int k = idx >> 6;
        w1[idx] = (k < 7) ? W1[idx] : 0.f;
    }
    __syncthreads();

    // ---- layer 1: [16x7] @ W1[7x64], bias folded into acc init, lrelu on writeback
    #pragma unroll
    for (int nt = 0; nt < 4; ++nt) {
        const int n = nt * 16 + l16;
        v8f acc = splat8(b1[n]);
        #pragma unroll
        for (int kb = 0; kb < 8; kb += 4) {
            const int k0 = kb + 2 * half;
            v2f a;  a.x = xa[l16 * 8 + k0];      a.y = xa[l16 * 8 + k0 + 1];
            v2f bf; bf.x = w1[k0 * 64 + n];      bf.y = w1[(k0 + 1) * 64 + n];
            acc = WMMA_F32(a, bf, acc);
        }
        #pragma unroll
        for (int r = 0; r < 8; ++r)
            h[(r + 8 * half) * 64 + n] = lrelu(acc[r]);
    }
    __syncthreads();

    // ---- layer 2: h[16x64] @ W2[64x64]; keep all 4 N-tiles resident in VGPRs
    v8f acc2[4];
    #pragma unroll
    for (int nt = 0; nt < 4; ++nt) acc2[nt] = splat8(b2[nt * 16 + l16]);
    for (int kb = 0; kb < 64; kb += 4) {
        const int k0 = kb + 2 * half;
        v2f a;  a.x = h[l16 * 64 + k0];  a.y = h[l16 * 64 + k0 + 1];
        #pragma unroll
        for (int nt = 0; nt < 4; ++nt) {
            const int n = nt * 16 + l16;
            v2f bf;  bf.x = W2[k0 * 64 + n];  bf.y = W2[(k0 + 1) * 64 + n];
            acc2[nt] = WMMA_F32(a, bf, acc2[nt]);
        }
    }
    __syncthreads();    // all layer-2 reads of h done before overwriting
    #pragma unroll
    for (int nt = 0; nt < 4; ++nt) {
        const int n = nt * 16 + l16;
        #pragma unroll
        for (int r = 0; r < 8; ++r)
            h[(r + 8 * half) * 64 + n] = lrelu(acc2[nt][r]);
    }
    __syncthreads();

    // ---- layer 3: h[16x64] @ W3[64x32] -> fx tile (no activation)
    v8f acc3[2];
    #pragma unroll
    for (int nt = 0; nt < 2; ++nt) acc3[nt] = splat8(b3[nt * 16 + l16]);
    for (int kb = 0; kb < 64; kb += 4) {
        const int k0 = kb + 2 * half;
        v2f a;  a.x = h[l16 * 64 + k0];  a.y = h[l16 * 64 + k0 + 1];
        #pragma unroll
        for (int nt = 0; nt < 2; ++nt) {
            const int n = nt * 16 + l16;
            v2f bf;  bf.x = W3[k0 * 32 + n];  bf.y = W3[(k0 + 1) * 32 + n];
            acc3[nt] = WMMA_F32(a, bf, acc3[nt]);
        }
    }
    #pragma unroll
    for (int nt = 0; nt < 2; ++nt) {
        const int n = nt * 16 + l16;
        #pragma unroll
        for (int r = 0; r < 8; ++r)
            fx[(row0 + r + 8 * half) * 32 + n] = acc3[nt][r];
    }
}

// -----------------------------------------------------------------------------
// Kernel 2: attention. score[b,i,j,h] = s_i + s_j + ba; softmax over j cancels
// the (s_i + ba) terms exactly, so att and the aggregation are independent of i:
// per batch we need one softmax over s_j per head, one HxD aggregation, one
// 128->32 projection, then broadcast the row to all N outputs. ~0.1% of the
// total work -> scalar VALU + LDS is the right tool here.
// -----------------------------------------------------------------------------
__global__ __launch_bounds__(256)
void gat_attn_kernel(const float* __restrict__ fx,
                     const float* __restrict__ Wa,   // [64,4]; rows 32..63 = col-node part
                     const float* __restrict__ Ws,   // [128,32]
                     const float* __restrict__ bs,   // [32]
                     float* __restrict__ out)        // [B,1024,32]
{
    const int b   = blockIdx.x;
    const int tid = threadIdx.x;
    __shared__ float sL[1024 * 4];   // s_j, then att weights, per head
    __shared__ float redM[4], redZ[4];
    __shared__ float agg[128];       // [H,D] head-major
    __shared__ float row[32];

    const float* fxb = fx + b * 1024 * 32;

    // s_j[j,h] = fx[b,j,:] @ Wa[32:,:]
    for (int j = tid; j < 1024; j += 256) {
        const float* f = fxb + j * 32;
        float s0 = 0.f, s1 = 0.f, s2 = 0.f, s3 = 0.f;
        #pragma unroll
        for (int d = 0; d < 32; ++d) {
            const float v = f[d];
            const float* w = Wa + (32 + d) * 4;
            s0 += v * w[0];  s1 += v * w[1];  s2 += v * w[2];  s3 += v * w[3];
        }
        sL[j * 4 + 0] = s0;  sL[j * 4 + 1] = s1;
        sL[j * 4 + 2] = s2;  sL[j * 4 + 3] = s3;
    }
    __syncthreads();

    // stable softmax stats per head
    if (tid < 4) {
        float m = -3.402823466e38f;
        for (int j = 0; j < 1024; ++j) m = fmaxf(m, sL[j * 4 + tid]);
        float z = 0.f;
        for (int j = 0; j < 1024; ++j) z += expf(sL[j * 4 + tid] - m);
        redM[tid] = m;  redZ[tid] = z;
    }
    __syncthreads();

    // normalize in place -> attention weights
    for (int j = tid; j < 1024; j += 256) {
        #pragma unroll
        for (int hh = 0; hh < 4; ++hh)
            sL[j * 4 + hh] = expf(sL[j * 4 + hh] - redM[hh]) / redZ[hh];
    }
    __syncthreads();

    // agg[h,d] = sum_j att[j,h] * fx[b,j,d]
    if (tid < 128) {
        const int hh = tid >> 5, d = tid & 31;
        float acc = 0.f;
        for (int j = 0; j < 1024; ++j)
            acc += sL[j * 4 + hh] * fxb[j * 32 + d];
        agg[tid] = acc;
    }
    __syncthreads();

    // row[d] = agg_cat @ Ws + bs
    if (tid < 32) {
        float r = bs[tid];
        #pragma unroll
        for (int k = 0; k < 128; ++k) r += agg[k] * Ws[k * 32 + tid];
        row[tid] = r;
    }
    __syncthreads();

    // broadcast to all N rows of this batch
    float* ob = out + b * 1024 * 32;
    for (int idx = tid; idx < 1024 * 32; idx += 256)
        ob[idx] = row[idx & 31];
}

extern "C" void kernel_launch(void* const* d_in, const int* in_sizes, int n_in,
                              void* d_out, int out_size, void* d_ws, size_t ws_size,
                              hipStream_t stream)
{
    const float* x  = (const float*)d_in[0];
    const float* W1 = (const float*)d_in[1];
    const float* b1 = (const float*)d_in[2];
    const float* W2 = (const float*)d_in[3];
    const float* b2 = (const float*)d_in[4];
    const float* W3 = (const float*)d_in[5];
    const float* b3 = (const float*)d_in[6];
    const float* Wa = (const float*)d_in[7];
    // d_in[8] = ba: cancels exactly inside softmax over j (constant along axis)
    const float* Ws = (const float*)d_in[9];
    const float* bs = (const float*)d_in[10];

    float* fx = (float*)d_ws;   // [8*1024, 32] fp32 = 1 MB scratch

    gat_mlp_kernel<<<(8 * 1024) / 16, 32, 0, stream>>>(x, W1, b1, W2, b2, W3, b3, fx);
    gat_attn_kernel<<<8, 256, 0, stream>>>(fx, Wa, Ws, bs, (float*)d_out);
}